// DenseGCN_13786845020601
// MI455X (gfx1250) — compile-verified
//
#include <hip/hip_runtime.h>

typedef __attribute__((ext_vector_type(2))) float v2f;
typedef __attribute__((ext_vector_type(8))) float v8f;

#define FT 128

// ---------------- degree / normalization ----------------

__global__ void __launch_bounds__(256)
k_init_deg(float* __restrict__ deg, int n) {
    int i = blockIdx.x * 256 + threadIdx.x;
    if (i < n) deg[i] = 1.0f;   // self-loop contributes 1
}

__global__ void __launch_bounds__(256)
k_edge_deg(const int* __restrict__ dst, float* __restrict__ deg, int E) {
    int e = blockIdx.x * 256 + threadIdx.x;
    if (e < E) unsafeAtomicAdd(&deg[dst[e]], 1.0f);
}

__global__ void __launch_bounds__(256)
k_rsqrt(float* __restrict__ deg, int n) {
    int i = blockIdx.x * 256 + threadIdx.x;
    if (i < n) deg[i] = rsqrtf(deg[i]);   // deg >= 1 always
}

// ---------------- weight transpose: Wt[col*K + k] = W[k*128 + col] ----------------

__global__ void __launch_bounds__(256)
k_transpose_w(const float* __restrict__ W, float* __restrict__ Wt, int K) {
    int i = blockIdx.x * 256 + threadIdx.x;
    if (i < K * FT) {
        int k = i >> 7, c = i & 127;
        Wt[(size_t)c * K + k] = W[i];
    }
}

// ---------------- fp32 WMMA GEMM: out[n x 128] = concat(A0,A1,A2)[n x K] @ W[K x 128] ----------------
// K is a template parameter so the B-tile stride (16*K floats) and all unrolled k-chunk
// offsets are compile-time constants -> every load is base + immediate offset, letting
// the scheduler batch loads ahead of the V_WMMA_F32_16X16X4_F32 stream.

template <int K>
__global__ void __launch_bounds__(256)
k_gemm_wmma(const float* __restrict__ A0, const float* __restrict__ A1,
            const float* __restrict__ A2, const float* __restrict__ Wt,
            const float* __restrict__ bias, float* __restrict__ out,
            int n, int fuse_bias_relu)
{
    const int lane = threadIdx.x & 31;
    const int half = lane >> 4;      // 0: K pair {0,1}, 1: K pair {2,3}
    const int l16  = lane & 15;
    const int wave = threadIdx.x >> 5;
    const int row0 = blockIdx.x * 128 + wave * 16;
    if (row0 >= n) return;           // wave-uniform exit, EXEC stays all-1s for live waves

    const v8f zero = {0.f, 0.f, 0.f, 0.f, 0.f, 0.f, 0.f, 0.f};
    v8f acc[8];
#pragma unroll
    for (int t = 0; t < 8; ++t) acc[t] = zero;

    int ar = row0 + l16;             // A-fragment row for this lane (clamped for tail)
    if (ar >= n) ar = n - 1;

    const float* segs[3] = {A0, A1, A2};
    const float* Wlane = Wt + (size_t)l16 * K + half * 2;   // column l16 of Wt, K-pair base

#pragma unroll
    for (int seg = 0; seg < (K >> 7); ++seg) {
        // per-segment bases; all inner-loop addresses are base + compile-time offset
        const float* Arow  = segs[seg] + (size_t)ar * FT + half * 2;
        const float* Bbase = Wlane + seg * 128;
#pragma unroll 4
        for (int kc = 0; kc < 32; ++kc) {
            // A 16x4 fp32 fragment: lane<16 holds K={0,1}, lane>=16 holds K={2,3}
            v2f a = *(const v2f*)(Arow + kc * 4);
#pragma unroll
            for (int t = 0; t < 8; ++t) {
                // B 4x16 fragment: two consecutive K values of column t*16+l16
                v2f b = *(const v2f*)(Bbase + kc * 4 + t * 16 * K);
                acc[t] = __builtin_amdgcn_wmma_f32_16x16x4_f32(
                    false, a, false, b, (short)0, acc[t], false, false);
            }
        }
    }

    // C/D layout: VGPR v, lanes 0-15 -> M=v, lanes 16-31 -> M=v+8, N=lane%16
#pragma unroll
    for (int v = 0; v < 8; ++v) {
        const int row = row0 + v + half * 8;
        if (row < n) {
#pragma unroll
            for (int t = 0; t < 8; ++t) {
                const int col = t * 16 + l16;
                float val = acc[t][v];
                if (fuse_bias_relu) val = fmaxf(val + bias[col], 0.0f);
                out[(size_t)row * FT + col] = val;
            }
        }
    }
}

// ---------------- aggregation ----------------

// h = tmp * dinv^2 + bias   (self-loop term + bias; full deterministic overwrite)
__global__ void __launch_bounds__(256)
k_agg_init(const float* __restrict__ tmp, const float* __restrict__ dinv,
           const float* __restrict__ bias, float* __restrict__ h, int n)
{
    int i = blockIdx.x * 256 + threadIdx.x;
    if (i < n * FT) {
        int row = i >> 7, col = i & 127;
        float di = dinv[row];
        h[i] = tmp[i] * di * di + bias[col];
    }
}

// one wave per edge, one float4 per lane (128 floats/row)
__global__ void __launch_bounds__(256)
k_edge_scatter(const int* __restrict__ src, const int* __restrict__ dst,
               const float* __restrict__ dinv, const float* __restrict__ tmp,
               float* __restrict__ h, int E)
{
    int idx  = blockIdx.x * 256 + threadIdx.x;
    int e    = idx >> 5;
    int lane = idx & 31;
    if (e >= E) return;
    int s = src[e], d = dst[e];
    float norm = dinv[s] * dinv[d];
    const float4 m = *(const float4*)(tmp + (size_t)s * FT + lane * 4);
    float* q = h + (size_t)d * FT + lane * 4;
    unsafeAtomicAdd(q + 0, m.x * norm);
    unsafeAtomicAdd(q + 1, m.y * norm);
    unsafeAtomicAdd(q + 2, m.z * norm);
    unsafeAtomicAdd(q + 3, m.w * norm);
}

__global__ void __launch_bounds__(256)
k_relu_inplace(float* __restrict__ h, int total4)
{
    int i = blockIdx.x * 256 + threadIdx.x;
    if (i < total4) {
        float4* p = (float4*)h + i;
        float4 v = *p;
        v.x = fmaxf(v.x, 0.f);
        v.y = fmaxf(v.y, 0.f);
        v.z = fmaxf(v.z, 0.f);
        v.w = fmaxf(v.w, 0.f);
        *p = v;
    }
}

// ---------------- orchestration ----------------

extern "C" void kernel_launch(void* const* d_in, const int* in_sizes, int n_in,
                              void* d_out, int out_size, void* d_ws, size_t ws_size,
                              hipStream_t stream)
{
    const float* x      = (const float*)d_in[0];
    const int*   eidx   = (const int*)  d_in[1];
    const float* proj_W = (const float*)d_in[2];
    const float* proj_b = (const float*)d_in[3];
    const float* W1     = (const float*)d_in[4];
    const float* b1     = (const float*)d_in[5];
    const float* W2     = (const float*)d_in[6];
    const float* b2     = (const float*)d_in[7];
    const float* W3     = (const float*)d_in[8];
    const float* b3     = (const float*)d_in[9];

    const int n = in_sizes[0] / FT;
    const int E = in_sizes[1] / 2;
    const int* src = eidx;
    const int* dst = eidx + E;

    float* ws = (float*)d_ws;
    const size_t nf = (size_t)n * FT;
    float* xp   = ws;                 // [n,128]
    float* h1   = ws + nf;            // [n,128]
    float* h2   = ws + 2 * nf;        // [n,128]
    float* tmp  = ws + 3 * nf;        // [n,128] pre-aggregation GEMM result
    float* dinv = ws + 4 * nf;        // [n]
    size_t off  = 4 * nf + (((size_t)n + 3) & ~(size_t)3);
    float* WtP  = ws + off;                    // 128x128 transposed proj_W
    float* Wt1  = WtP + (size_t)128 * FT;      // 128x128 transposed W1
    float* Wt2  = Wt1 + (size_t)128 * FT;      // 128x256 transposed W2
    float* Wt3  = Wt2 + (size_t)256 * FT;      // 128x384 transposed W3
    float* hout = (float*)d_out;               // h3 [n,128]

    const int gElem  = (n * FT + 255) / 256;
    const int gElem4 = (n * FT / 4 + 255) / 256;
    const int gNode  = (n + 255) / 256;
    const int gEdge  = (E + 255) / 256;
    const int gScat  = (int)(((size_t)E * 32 + 255) / 256);
    const int gGemm  = (n + 127) / 128;

    // weight transposes (tiny)
    k_transpose_w<<<(128 * FT + 255) / 256, 256, 0, stream>>>(proj_W, WtP, 128);
    k_transpose_w<<<(128 * FT + 255) / 256, 256, 0, stream>>>(W1, Wt1, 128);
    k_transpose_w<<<(256 * FT + 255) / 256, 256, 0, stream>>>(W2, Wt2, 256);
    k_transpose_w<<<(384 * FT + 255) / 256, 256, 0, stream>>>(W3, Wt3, 384);

    // symmetric normalization coefficients (shared by all three convs)
    k_init_deg<<<gNode, 256, 0, stream>>>(dinv, n);
    k_edge_deg<<<gEdge, 256, 0, stream>>>(dst, dinv, E);
    k_rsqrt  <<<gNode, 256, 0, stream>>>(dinv, n);

    // x_p = relu(x @ proj_W + proj_b)
    k_gemm_wmma<128><<<gGemm, 256, 0, stream>>>(x, nullptr, nullptr, WtP, proj_b, xp, n, 1);

    // h1 = relu(gcn_conv(x, W1, b1))
    k_gemm_wmma<128><<<gGemm, 256, 0, stream>>>(x, nullptr, nullptr, Wt1, nullptr, tmp, n, 0);
    k_agg_init    <<<gElem, 256, 0, stream>>>(tmp, dinv, b1, h1, n);
    k_edge_scatter<<<gScat, 256, 0, stream>>>(src, dst, dinv, tmp, h1, E);
    k_relu_inplace<<<gElem4, 256, 0, stream>>>(h1, n * FT / 4);

    // h2 = relu(gcn_conv(concat[x_p,h1], W2, b2))
    k_gemm_wmma<256><<<gGemm, 256, 0, stream>>>(xp, h1, nullptr, Wt2, nullptr, tmp, n, 0);
    k_agg_init    <<<gElem, 256, 0, stream>>>(tmp, dinv, b2, h2, n);
    k_edge_scatter<<<gScat, 256, 0, stream>>>(src, dst, dinv, tmp, h2, E);
    k_relu_inplace<<<gElem4, 256, 0, stream>>>(h2, n * FT / 4);

    // h3 = relu(gcn_conv(concat[x_p,h1,h2], W3, b3)) -> d_out
    k_gemm_wmma<384><<<gGemm, 256, 0, stream>>>(xp, h1, h2, Wt3, nullptr, tmp, n, 0);
    k_agg_init    <<<gElem, 256, 0, stream>>>(tmp, dinv, b3, hout, n);
    k_edge_scatter<<<gScat, 256, 0, stream>>>(src, dst, dinv, tmp, hout, E);
    k_relu_inplace<<<gElem4, 256, 0, stream>>>(hout, n * FT / 4);
}